// RelationalGraphSAGELayer_37752762531901
// MI455X (gfx1250) — compile-verified
//
#include <hip/hip_runtime.h>
#include <cstdint>
#include <cstddef>

// ---------------------------------------------------------------------------
// RelationalGraphSAGE layer for MI455X (gfx1250, wave32).
//
// Rewrite: scatter-add RAW masked features (linear op commutes with
// segment_sum), then one dense [N,128]x[128,128] GEMM per table using
// V_WMMA_F32_16X16X4_F32 (exact f32). Memory-bound (~2 GB @ 23.3 TB/s),
// GEMMs are only ~39 GFLOP after the rewrite.
// ---------------------------------------------------------------------------

typedef float v2f __attribute__((ext_vector_type(2)));
typedef float v8f __attribute__((ext_vector_type(8)));

static constexpr int FD   = 128;      // feature dim
static constexpr int NA   = 200000;
static constexpr int NB   = 400000;
static constexpr int EP2C = 800000;
static constexpr int EC2P = 800000;
static constexpr int WAVES_PER_BLOCK = 4;
static constexpr int XSTRIDE = 132;   // 128 + 4 pad: conflict-free frag reads

// ---------------------------------------------------------------------------
// Phase 1: one wave per edge; lane handles 4 contiguous columns.
// agg[dst] += mask * feats[src]; deg[dst] += mask.
// ---------------------------------------------------------------------------
__global__ __launch_bounds__(256) void edge_scatter_kernel(
    const float* __restrict__ feats_src,
    const int*   __restrict__ src,
    const int*   __restrict__ dst,
    const float* __restrict__ u,
    float*       __restrict__ agg,
    float*       __restrict__ deg,
    int nEdges)
{
    int gid  = blockIdx.x * blockDim.x + threadIdx.x;
    int e    = gid >> 5;
    int lane = gid & 31;
    if (e >= nEdges) return;
    if (!(u[e] < 0.5f)) return;            // update_keep_prob = 0.5

    int s = src[e];
    int d = dst[e];
    const float4* sp = reinterpret_cast<const float4*>(feats_src + (size_t)s * FD);
    float4 v = sp[lane];
    float* ap = agg + (size_t)d * FD + lane * 4;
    unsafeAtomicAdd(ap + 0, v.x);
    unsafeAtomicAdd(ap + 1, v.y);
    unsafeAtomicAdd(ap + 2, v.z);
    unsafeAtomicAdd(ap + 3, v.w);
    if (lane == 0) unsafeAtomicAdd(deg + d, 1.0f);
}

// ---------------------------------------------------------------------------
// Phase 2: one wave per 16-row tile. 8 x v8f accumulators hold the 16x128
// output tile. Two K=128 WMMA passes: feats@Wself^T then
// (agg/deg)@(fmask (.) Wedge)^T. Then bias, LayerNorm (half-wave shuffle
// reduction), SELU residual, masked column write-back — all in registers.
// inout = raw agg on entry (from phase 1), final table on exit (in place).
// ---------------------------------------------------------------------------
__global__ __launch_bounds__(32 * WAVES_PER_BLOCK) void table_update_kernel(
    const float* __restrict__ feats,
    float*       __restrict__ inout,
    const float* __restrict__ deg,
    const float* __restrict__ Wself, const float* __restrict__ bself,
    const float* __restrict__ Wedge, const float* __restrict__ bedge,
    const float* __restrict__ lnw,   const float* __restrict__ lnb,
    const float* __restrict__ ufeat,
    int nTiles)
{
    __shared__ __align__(16) float Xlds[WAVES_PER_BLOCK][16 * XSTRIDE];

    const int wid  = threadIdx.x >> 5;
    const int lane = threadIdx.x & 31;
    const int tile = blockIdx.x * WAVES_PER_BLOCK + wid;
    if (tile >= nTiles) return;            // no block barriers used -> safe

    const int hh  = lane >> 4;             // half-wave: 0 or 1
    const int l16 = lane & 15;
    float* X = &Xlds[wid][0];
    const size_t rowbase = (size_t)tile * 16;

    v8f acc[8];
    const v8f vzero = {0.f, 0.f, 0.f, 0.f, 0.f, 0.f, 0.f, 0.f};
#pragma unroll
    for (int t = 0; t < 8; ++t) acc[t] = vzero;

    // ---- stage feats tile into LDS (coalesced row loads) ----
#pragma unroll 4
    for (int r = 0; r < 16; ++r) {
        const float4* fp = reinterpret_cast<const float4*>(feats + (rowbase + r) * FD);
        *reinterpret_cast<float4*>(X + r * XSTRIDE + lane * 4) = fp[lane];
    }
    asm volatile("s_wait_dscnt 0x0" ::: "memory");

    // ---- GEMM 1: acc += featsTile @ Wself^T ----
    // A frag: lane l -> (M=l16, K=k+2*hh + v). B frag: lane l -> (K, N=l16).
    for (int k = 0; k < FD; k += 4) {
        v2f a = *reinterpret_cast<const v2f*>(X + l16 * XSTRIDE + k + 2 * hh);
#pragma unroll
        for (int t = 0; t < 8; ++t) {
            v2f b = *reinterpret_cast<const v2f*>(
                Wself + (size_t)(t * 16 + l16) * FD + k + 2 * hh);
            acc[t] = __builtin_amdgcn_wmma_f32_16x16x4_f32(
                false, a, false, b, (short)0, acc[t], false, false);
        }
    }

    // ---- stage agg tile, pre-scaled by 1/max(deg,1); record deg>0 flag ----
    float sd8[8];
#pragma unroll 4
    for (int r = 0; r < 16; ++r) {
        size_t row = rowbase + r;
        float dg = deg[row];
        float rs = 1.0f / fmaxf(dg, 1.0f);
        if ((r >> 3) == hh) sd8[r & 7] = (dg > 0.0f) ? 1.0f : 0.0f;
        const float4* ap = reinterpret_cast<const float4*>(inout + row * FD);
        float4 v = ap[lane];
        v.x *= rs; v.y *= rs; v.z *= rs; v.w *= rs;
        *reinterpret_cast<float4*>(X + r * XSTRIDE + lane * 4) = v;
    }
    asm volatile("s_wait_dscnt 0x0" ::: "memory");

    // per-output-column feature mask (feature_keep_prob = 0.3)
    float fmv[8];
#pragma unroll
    for (int t = 0; t < 8; ++t)
        fmv[t] = (ufeat[t * 16 + l16] < 0.3f) ? 1.0f : 0.0f;

    // ---- GEMM 2: acc += (agg/deg) @ (fmask (.)rows Wedge)^T ----
    for (int k = 0; k < FD; k += 4) {
        v2f a = *reinterpret_cast<const v2f*>(X + l16 * XSTRIDE + k + 2 * hh);
#pragma unroll
        for (int t = 0; t < 8; ++t) {
            v2f b = *reinterpret_cast<const v2f*>(
                Wedge + (size_t)(t * 16 + l16) * FD + k + 2 * hh);
            b = b * fmv[t];                 // fold column mask into B operand
            acc[t] = __builtin_amdgcn_wmma_f32_16x16x4_f32(
                false, a, false, b, (short)0, acc[t], false, false);
        }
    }

    // ---- epilogue: biases ----
    float bsv[8], bev[8], lwv[8], lbv[8];
#pragma unroll
    for (int t = 0; t < 8; ++t) {
        int n = t * 16 + l16;
        bsv[t] = bself[n];
        bev[t] = bedge[n];
        lwv[t] = lnw[n];
        lbv[t] = lnb[n];
    }
#pragma unroll
    for (int t = 0; t < 8; ++t) {
#pragma unroll
        for (int r = 0; r < 8; ++r)
            acc[t][r] += bsv[t] + sd8[r] * fmv[t] * bev[t];
    }

    // ---- LayerNorm stats: row = r + 8*hh, reduce over 16-lane half ----
    float mu[8], rstd[8];
#pragma unroll
    for (int r = 0; r < 8; ++r) {
        float s = 0.f, q = 0.f;
#pragma unroll
        for (int t = 0; t < 8; ++t) { float x = acc[t][r]; s += x; q += x * x; }
#pragma unroll
        for (int m = 1; m <= 8; m <<= 1) {
            s += __shfl_xor(s, m, 32);
            q += __shfl_xor(q, m, 32);
        }
        float mean = s * (1.0f / 128.0f);
        float var  = q * (1.0f / 128.0f) - mean * mean;
        mu[r]   = mean;
        rstd[r] = rsqrtf(var + 1e-5f);
    }

    // ---- LN affine + SELU residual + masked write-back ----
    const float SELU_SCALE = 1.0507009873554805f;
    const float SELU_ALPHA = 1.6732632423543772f;
#pragma unroll
    for (int t = 0; t < 8; ++t) {
        int n = t * 16 + l16;
#pragma unroll
        for (int r = 0; r < 8; ++r) {
            size_t row = rowbase + (size_t)(hh * 8 + r);
            float f = feats[row * FD + n];
            float x = (acc[t][r] - mu[r]) * rstd[r] * lwv[t] + lbv[t] + f;
            float sv = SELU_SCALE * (x > 0.f ? x : SELU_ALPHA * (__expf(x) - 1.0f));
            inout[row * FD + n] = (fmv[t] > 0.5f) ? sv : f;
        }
    }
}

// ---------------------------------------------------------------------------
extern "C" void kernel_launch(void* const* d_in, const int* in_sizes, int n_in,
                              void* d_out, int out_size, void* d_ws, size_t ws_size,
                              hipStream_t stream) {
    (void)in_sizes; (void)n_in; (void)ws_size;

    const float* feats_A  = (const float*)d_in[0];
    const float* feats_B  = (const float*)d_in[1];
    const float* W_p2c    = (const float*)d_in[2];
    const float* b_p2c    = (const float*)d_in[3];
    const float* W_c2p    = (const float*)d_in[4];
    const float* b_c2p    = (const float*)d_in[5];
    const float* Wself_A  = (const float*)d_in[6];
    const float* bself_A  = (const float*)d_in[7];
    const float* lnw_A    = (const float*)d_in[8];
    const float* lnb_A    = (const float*)d_in[9];
    const float* Wself_B  = (const float*)d_in[10];
    const float* bself_B  = (const float*)d_in[11];
    const float* lnw_B    = (const float*)d_in[12];
    const float* lnb_B    = (const float*)d_in[13];
    const float* u_p2c    = (const float*)d_in[14];
    const float* u_c2p    = (const float*)d_in[15];
    const float* u_feat_A = (const float*)d_in[16];
    const float* u_feat_B = (const float*)d_in[17];
    const int*   src_p2c  = (const int*)d_in[18];
    const int*   dst_p2c  = (const int*)d_in[19];
    const int*   src_c2p  = (const int*)d_in[20];
    const int*   dst_c2p  = (const int*)d_in[21];

    float* outA = (float*)d_out;                      // 200000 x 128
    float* outB = outA + (size_t)NA * FD;             // 400000 x 128
    float* degB = (float*)d_ws;                       // 400000 floats
    float* degA = degB + NB;                          // 200000 floats

    // d_out doubles as the raw-feature aggregation buffer; zero both.
    hipMemsetAsync(d_out, 0, (size_t)out_size * sizeof(float), stream);
    hipMemsetAsync(d_ws, 0, (size_t)(NA + NB) * sizeof(float), stream);

    // Phase 1: edge scatter (A->B with p2c edges; B->A with c2p edges)
    {
        long long th = (long long)EP2C * 32;
        edge_scatter_kernel<<<(unsigned)((th + 255) / 256), 256, 0, stream>>>(
            feats_A, src_p2c, dst_p2c, u_p2c, outB, degB, EP2C);
    }
    {
        long long th = (long long)EC2P * 32;
        edge_scatter_kernel<<<(unsigned)((th + 255) / 256), 256, 0, stream>>>(
            feats_B, src_c2p, dst_c2p, u_c2p, outA, degA, EC2P);
    }

    // Phase 2: per-table fused GEMM + LN + SELU (in place on d_out regions).
    // Table A receives via c2p edges (W_c2p), table B via p2c edges (W_p2c).
    {
        int tiles = NA / 16;   // 12500, divisible by WAVES_PER_BLOCK
        table_update_kernel<<<tiles / WAVES_PER_BLOCK, 32 * WAVES_PER_BLOCK, 0, stream>>>(
            feats_A, outA, degA, Wself_A, bself_A, W_c2p, b_c2p,
            lnw_A, lnb_A, u_feat_A, tiles);
    }
    {
        int tiles = NB / 16;   // 25000
        table_update_kernel<<<tiles / WAVES_PER_BLOCK, 32 * WAVES_PER_BLOCK, 0, stream>>>(
            feats_B, outB, degB, Wself_B, bself_B, W_p2c, b_p2c,
            lnw_B, lnb_B, u_feat_B, tiles);
    }
}